// TemporalGNN_5566277616471
// MI455X (gfx1250) — compile-verified
//
#include <hip/hip_runtime.h>
#include <hip/hip_bf16.h>

#define NNODES 30000
#define NEDGES 480000
#define BATCH  4
#define FEAT   32
#define TSTEPS 12
#define OUTF   64

typedef float v2f __attribute__((ext_vector_type(2)));
typedef float v8f __attribute__((ext_vector_type(8)));

// ---------------------------------------------------------------------------
// Fold weights: W2[f][0:64] = Wg_z@Wl_z_top, W2[f][64:128] = Wg_h@Wl_h_top
// c2 = folded biases, probs = softmax(att). One 256-thread block.
// ---------------------------------------------------------------------------
__global__ void prep_kernel(const float* __restrict__ Wg_z, const float* __restrict__ bg_z,
                            const float* __restrict__ Wg_h, const float* __restrict__ bg_h,
                            const float* __restrict__ Wl_z, const float* __restrict__ bl_z,
                            const float* __restrict__ Wl_h, const float* __restrict__ bl_h,
                            const float* __restrict__ att,
                            float* __restrict__ W2, float* __restrict__ c2,
                            float* __restrict__ probs) {
  int tid = threadIdx.x;
  for (int idx = tid; idx < FEAT * 128; idx += 256) {
    int f = idx >> 7, col = idx & 127;
    int gate = col >> 6, o = col & 63;
    const float* Wg = gate ? Wg_h : Wg_z;
    const float* Wl = gate ? Wl_h : Wl_z;
    float s = 0.f;
    for (int k = 0; k < 64; ++k) s += Wg[f * 64 + k] * Wl[k * 64 + o];
    W2[idx] = s;
  }
  if (tid < 128) {
    int gate = tid >> 6, o = tid & 63;
    const float* bg = gate ? bg_h : bg_z;
    const float* Wl = gate ? Wl_h : Wl_z;
    const float* bl = gate ? bl_h : bl_z;
    float s = bl[o];
    for (int k = 0; k < 64; ++k) s += bg[k] * Wl[k * 64 + o];
    c2[tid] = s;
  }
  if (tid == 0) {
    float mx = att[0];
    for (int t = 1; t < TSTEPS; ++t) mx = fmaxf(mx, att[t]);
    float e[TSTEPS]; float sum = 0.f;
    for (int t = 0; t < TSTEPS; ++t) { e[t] = __expf(att[t] - mx); sum += e[t]; }
    for (int t = 0; t < TSTEPS; ++t) probs[t] = e[t] / sum;
  }
}

// ---------------------------------------------------------------------------
// Degree / normalization:  deg = in_degree(dst) + 1 ; dis = rsqrt(deg)
// ---------------------------------------------------------------------------
__global__ void deg_init(float* dis) {
  int i = blockIdx.x * 256 + threadIdx.x;
  if (i < NNODES) dis[i] = 1.0f;
}
__global__ void deg_edges(const int* __restrict__ dst, float* __restrict__ dis) {
  int e = blockIdx.x * 256 + threadIdx.x;
  if (e < NEDGES) atomicAdd(&dis[dst[e]], 1.0f);
}
__global__ void deg_finish(float* dis) {
  int i = blockIdx.x * 256 + threadIdx.x;
  if (i < NNODES) dis[i] = rsqrtf(dis[i]);
}

__global__ void zero_kernel(float* __restrict__ p, int n) {
  int i = blockIdx.x * 256 + threadIdx.x;
  if (i < n) p[i] = 0.f;
}

// ---------------------------------------------------------------------------
// For fixed (b,t): Xt[n,f] = X[b,n,f,t] (contiguous copy for gathers)
//                  G[n,f]  = dis[n]^2 * Xt[n,f]   (self-loop term)
// ---------------------------------------------------------------------------
__global__ void self_kernel(const float* __restrict__ X, const float* __restrict__ dis,
                            float* __restrict__ Xt, float* __restrict__ G, int b, int t) {
  int i = blockIdx.x * 256 + threadIdx.x;
  if (i >= NNODES * FEAT) return;
  int n = i >> 5, f = i & 31;
  float x = X[((size_t)(b * NNODES + n) * FEAT + f) * TSTEPS + t];
  float d = dis[n];
  Xt[i] = x;
  G[i] = d * d * x;
}

// ---------------------------------------------------------------------------
// Edge scatter: G[dst,f] += dis[src]*dis[dst] * Xt[src,f]  (one wave per edge)
// ---------------------------------------------------------------------------
__global__ void edge_kernel(const int* __restrict__ src, const int* __restrict__ dst,
                            const float* __restrict__ dis, const float* __restrict__ Xt,
                            float* __restrict__ G) {
  int gid = blockIdx.x * 256 + threadIdx.x;
  int e = gid >> 5;
  int f = gid & 31;
  if (e >= NEDGES) return;
  int s = src[e], d = dst[e];
  float w = dis[s] * dis[d];
  atomicAdd(&G[d * FEAT + f], w * Xt[s * FEAT + f]);
}

// ---------------------------------------------------------------------------
// Gate kernel (WMMA, f32 full precision):
//   [G(16x32)] @ [W2(32x128)] -> Gz|Gh ; Hn = (1-sigmoid(Gz))*tanh(Gh)
//   acc[row, 0:64] += p_t * Hn
// One wave per 16-row tile; 8 waves per block; W2 staged in LDS (stride 136).
// ---------------------------------------------------------------------------
#define LDS_STRIDE 136

__global__ void __launch_bounds__(256) gate_kernel(const float* __restrict__ G,
                                                   const float* __restrict__ W2,
                                                   const float* __restrict__ c2,
                                                   const float* __restrict__ probs,
                                                   float* __restrict__ acc, int t) {
  __shared__ float ldsW[FEAT * LDS_STRIDE];
  __shared__ float ldsC[128];
  for (int i = threadIdx.x; i < FEAT * 128; i += 256) {
    int f = i >> 7, col = i & 127;
    ldsW[f * LDS_STRIDE + col] = W2[i];
  }
  if (threadIdx.x < 128) ldsC[threadIdx.x] = c2[threadIdx.x];
  __syncthreads();

  float p = probs[t];
  int lane = threadIdx.x & 31;
  int wave = threadIdx.x >> 5;
  int row0 = (blockIdx.x * 8 + wave) * 16;
  if (row0 >= NNODES) return;

  int hi = lane >> 4;       // which half-wave (K pair selector)
  int m  = lane & 15;       // M for A, N for B/C

  int arow = row0 + m;
  if (arow >= NNODES) arow = NNODES - 1;  // safe clamp (keeps EXEC full)

  // A fragments: step s covers K = 4s..4s+3 ; lane supplies K = 4s+2*hi+{0,1}
  v2f a[8];
  const float* gp = G + (size_t)arow * FEAT + 2 * hi;
#pragma unroll
  for (int s = 0; s < 8; ++s) {
    a[s].x = gp[4 * s];
    a[s].y = gp[4 * s + 1];
  }

  v8f cz[4], ch[4];
#pragma unroll
  for (int c = 0; c < 4; ++c) {
    v8f az = {0.f, 0.f, 0.f, 0.f, 0.f, 0.f, 0.f, 0.f};
    v8f ah = {0.f, 0.f, 0.f, 0.f, 0.f, 0.f, 0.f, 0.f};
#pragma unroll
    for (int s = 0; s < 8; ++s) {
      int krow = 4 * s + 2 * hi;
      v2f bz, bh;
      bz.x = ldsW[krow * LDS_STRIDE + 16 * c + m];
      bz.y = ldsW[(krow + 1) * LDS_STRIDE + 16 * c + m];
      bh.x = ldsW[krow * LDS_STRIDE + 64 + 16 * c + m];
      bh.y = ldsW[(krow + 1) * LDS_STRIDE + 64 + 16 * c + m];
      az = __builtin_amdgcn_wmma_f32_16x16x4_f32(false, a[s], false, bz, (short)0, az,
                                                 false, false);
      ah = __builtin_amdgcn_wmma_f32_16x16x4_f32(false, a[s], false, bh, (short)0, ah,
                                                 false, false);
    }
    cz[c] = az;
    ch[c] = ah;
  }

  // C/D layout: element (row0 + v + 8*hi, 16c + m) in component v
#pragma unroll
  for (int c = 0; c < 4; ++c) {
    int col = 16 * c + m;
    float biasz = ldsC[col];
    float biash = ldsC[64 + col];
#pragma unroll
    for (int v = 0; v < 8; ++v) {
      int row = row0 + v + 8 * hi;
      if (row < NNODES) {
        float z = cz[c][v] + biasz;
        float h = ch[c][v] + biash;
        float sig = 1.0f / (1.0f + __expf(-z));
        float hn = (1.0f - sig) * tanhf(h);
        acc[(size_t)row * OUTF + col] += p * hn;
      }
    }
  }
}

// ---------------------------------------------------------------------------
// out[row,:] = relu(acc[row,:]) @ W_out + b_out      (row = b*N+n)
// ---------------------------------------------------------------------------
__global__ void out_kernel(const float* __restrict__ acc, const float* __restrict__ W_out,
                           const float* __restrict__ b_out, float* __restrict__ out) {
  int row = blockIdx.x * 256 + threadIdx.x;
  if (row >= BATCH * NNODES) return;
  const float* ar = acc + (size_t)row * OUTF;
  float r[OUTF];
#pragma unroll
  for (int k = 0; k < OUTF; ++k) r[k] = fmaxf(ar[k], 0.f);
  for (int t = 0; t < TSTEPS; ++t) {
    float s = b_out[t];
#pragma unroll
    for (int k = 0; k < OUTF; ++k) s += r[k] * W_out[k * TSTEPS + t];
    out[(size_t)row * TSTEPS + t] = s;
  }
}

// ---------------------------------------------------------------------------
extern "C" void kernel_launch(void* const* d_in, const int* in_sizes, int n_in,
                              void* d_out, int out_size, void* d_ws, size_t ws_size,
                              hipStream_t stream) {
  (void)in_sizes; (void)n_in; (void)out_size; (void)ws_size;
  const float* X    = (const float*)d_in[0];
  const int*   ei   = (const int*)d_in[1];
  const int*   src  = ei;
  const int*   dst  = ei + NEDGES;
  const float* Wg_z = (const float*)d_in[2];
  const float* bg_z = (const float*)d_in[3];
  // d_in[4], d_in[5] (Wg_r, bg_r) and d_in[10], d_in[11] (Wl_r, bl_r) are dead:
  // H0 == 0 in the reference, so the R gate never affects the output.
  const float* Wg_h = (const float*)d_in[6];
  const float* bg_h = (const float*)d_in[7];
  const float* Wl_z = (const float*)d_in[8];
  const float* bl_z = (const float*)d_in[9];
  const float* Wl_h = (const float*)d_in[12];
  const float* bl_h = (const float*)d_in[13];
  const float* att  = (const float*)d_in[14];
  const float* Wout = (const float*)d_in[15];
  const float* bout = (const float*)d_in[16];
  float* out = (float*)d_out;

  float* ws    = (float*)d_ws;
  float* probs = ws;                 // 12   (pad to 64)
  float* c2    = ws + 64;            // 128
  float* W2    = ws + 256;           // 4096
  float* dis   = ws + 4352;          // 30000
  float* Xt    = ws + 34816;         // 960000
  float* G     = ws + 994816;        // 960000
  float* acc   = ws + 1954816;       // 7680000   (total ~37 MB)

  prep_kernel<<<1, 256, 0, stream>>>(Wg_z, bg_z, Wg_h, bg_h, Wl_z, bl_z, Wl_h, bl_h,
                                     att, W2, c2, probs);
  deg_init<<<(NNODES + 255) / 256, 256, 0, stream>>>(dis);
  deg_edges<<<(NEDGES + 255) / 256, 256, 0, stream>>>(dst, dis);
  deg_finish<<<(NNODES + 255) / 256, 256, 0, stream>>>(dis);
  zero_kernel<<<(BATCH * NNODES * OUTF + 255) / 256, 256, 0, stream>>>(
      acc, BATCH * NNODES * OUTF);

  for (int b = 0; b < BATCH; ++b) {
    for (int t = 0; t < TSTEPS; ++t) {
      self_kernel<<<(NNODES * FEAT + 255) / 256, 256, 0, stream>>>(X, dis, Xt, G, b, t);
      edge_kernel<<<(NEDGES * 32 + 255) / 256, 256, 0, stream>>>(src, dst, dis, Xt, G);
      gate_kernel<<<(NNODES / 16 + 7) / 8, 256, 0, stream>>>(
          G, W2, c2, probs, acc + (size_t)b * NNODES * OUTF, t);
    }
  }
  out_kernel<<<(BATCH * NNODES + 255) / 256, 256, 0, stream>>>(acc, Wout, bout, out);
}